// EdgeAttributePredictorConvMP_36197984370756
// MI455X (gfx1250) — compile-verified
//
#include <hip/hip_runtime.h>
#include <hip/hip_bf16.h>
#include <stdint.h>

typedef __attribute__((ext_vector_type(2))) float v2f;
typedef __attribute__((ext_vector_type(8))) float v8f;

// ---------------------------------------------------------------------------
// Generic fp32 WMMA GEMM: D[M,N] = A[M,K] @ B[K,N] (+bias[N]) (+Cadd[M,N]) (relu?)
// Block tile 128x64, K-step 16; 8 waves, each wave owns a 32x32 sub-tile
// (2x2 grid of 16x16 WMMA fragments, 4 k-steps of V_WMMA_F32_16X16X4_F32).
// Double-buffered LDS staging via GLOBAL_LOAD_ASYNC_TO_LDS_B128 (ASYNCcnt),
// prefetching tile k+1 while WMMAs consume tile k. Ragged tiles use a guarded
// synchronous fallback.
// ---------------------------------------------------------------------------
__global__ __launch_bounds__(256) void gemm_wmma_f32(
    const float* __restrict__ A, const float* __restrict__ B,
    const float* __restrict__ bias, const float* __restrict__ Cadd,
    float* __restrict__ D, int M, int N, int K, int doRelu)
{
  __shared__ float As[2][128 * 20];   // stride 20 floats: conflict-free, 16B-aligned rows
  __shared__ float Bs[2][16 * 68];    // stride 68 floats

  const int tid   = threadIdx.x;
  const int wave  = tid >> 5;
  const int lane  = tid & 31;
  const int wm    = wave >> 1;     // 0..3  (M direction)
  const int wn    = wave & 1;      // 0..1  (N direction)
  const int lrow  = lane & 15;
  const int khalf = lane >> 4;     // lanes 0-15 hold K{0,1}, 16-31 hold K{2,3}

  const int m0 = blockIdx.y * 128;
  const int n0 = blockIdx.x * 64;
  const bool full = (m0 + 128 <= M) && (n0 + 64 <= N);   // block-uniform
  const int nk = K >> 4;                                  // K is a multiple of 16

  v8f acc[2][2];
  const v8f vzero = {0.f, 0.f, 0.f, 0.f, 0.f, 0.f, 0.f, 0.f};
#pragma unroll
  for (int i = 0; i < 2; ++i)
#pragma unroll
    for (int j = 0; j < 2; ++j) acc[i][j] = vzero;

  auto load_tile = [&](int buf, int k0) {
    if (full) {
      // A block 128x16: 2 async B128 per thread (global -> LDS, no VGPR staging)
#pragma unroll
      for (int it = 0; it < 2; ++it) {
        int i  = tid + it * 256;          // 0..511 float4 slots
        int r  = i >> 2;
        int c4 = i & 3;
        const float* g = A + (size_t)(m0 + r) * K + k0 + c4 * 4;
        unsigned l = (unsigned)(uintptr_t)(&As[buf][r * 20 + c4 * 4]);
        asm volatile("global_load_async_to_lds_b128 %0, %1, off"
                     :: "v"(l), "v"(g) : "memory");
      }
      // B block 16x64: 1 async B128 per thread
      {
        int r  = tid >> 4;
        int c4 = tid & 15;
        const float* g = B + (size_t)(k0 + r) * N + n0 + c4 * 4;
        unsigned l = (unsigned)(uintptr_t)(&Bs[buf][r * 68 + c4 * 4]);
        asm volatile("global_load_async_to_lds_b128 %0, %1, off"
                     :: "v"(l), "v"(g) : "memory");
      }
    } else {
      // Guarded synchronous fallback for ragged tiles
#pragma unroll
      for (int it = 0; it < 2; ++it) {
        int i  = tid + it * 256;
        int r  = i >> 2;
        int c4 = i & 3;
        int gm = m0 + r;
        float4 v = make_float4(0.f, 0.f, 0.f, 0.f);
        if (gm < M) v = *reinterpret_cast<const float4*>(A + (size_t)gm * K + k0 + c4 * 4);
        *reinterpret_cast<float4*>(&As[buf][r * 20 + c4 * 4]) = v;
      }
#pragma unroll
      for (int it = 0; it < 4; ++it) {
        int i  = tid + it * 256;
        int r  = i >> 6;
        int c  = i & 63;
        int gn = n0 + c;
        float v = 0.f;
        if (gn < N) v = B[(size_t)(k0 + r) * N + gn];
        Bs[buf][r * 68 + c] = v;
      }
    }
  };

  load_tile(0, 0);
  asm volatile("s_wait_asynccnt 0x0" ::: "memory");   // barriers don't wait ASYNCcnt
  __syncthreads();

  for (int t = 0; t < nk; ++t) {
    const int cur = t & 1;
    if (t + 1 < nk) load_tile(cur ^ 1, (t + 1) * 16);   // async prefetch next tile

    const float* Ac = &As[cur][0];
    const float* Bc = &Bs[cur][0];
#pragma unroll
    for (int kk = 0; kk < 16; kk += 4) {
      const int acol = kk + khalf * 2;
      v2f a0, a1, b0, b1;
      a0.x = Ac[(wm * 32 + lrow) * 20 + acol];
      a0.y = Ac[(wm * 32 + lrow) * 20 + acol + 1];
      a1.x = Ac[(wm * 32 + 16 + lrow) * 20 + acol];
      a1.y = Ac[(wm * 32 + 16 + lrow) * 20 + acol + 1];
      b0.x = Bc[acol * 68 + wn * 32 + lrow];
      b0.y = Bc[(acol + 1) * 68 + wn * 32 + lrow];
      b1.x = Bc[acol * 68 + wn * 32 + 16 + lrow];
      b1.y = Bc[(acol + 1) * 68 + wn * 32 + 16 + lrow];
      acc[0][0] = __builtin_amdgcn_wmma_f32_16x16x4_f32(false, a0, false, b0, (short)0, acc[0][0], false, false);
      acc[0][1] = __builtin_amdgcn_wmma_f32_16x16x4_f32(false, a0, false, b1, (short)0, acc[0][1], false, false);
      acc[1][0] = __builtin_amdgcn_wmma_f32_16x16x4_f32(false, a1, false, b0, (short)0, acc[1][0], false, false);
      acc[1][1] = __builtin_amdgcn_wmma_f32_16x16x4_f32(false, a1, false, b1, (short)0, acc[1][1], false, false);
    }

    asm volatile("s_wait_asynccnt 0x0" ::: "memory");   // next tile landed in LDS
    __syncthreads();                                    // all waves done with cur
  }

  // Epilogue: C/D layout — VGPR g: row = g + 8*khalf; col = lane&15
  const int nbase = n0 + wn * 32;
  const int mbase = m0 + wm * 32;
#pragma unroll
  for (int mi = 0; mi < 2; ++mi) {
#pragma unroll
    for (int ni = 0; ni < 2; ++ni) {
      int gn = nbase + ni * 16 + lrow;
      float bv = (bias != nullptr && gn < N) ? bias[gn] : 0.f;
#pragma unroll
      for (int g = 0; g < 8; ++g) {
        int gm = mbase + mi * 16 + khalf * 8 + g;
        if (gm < M && gn < N) {
          float v = acc[mi][ni][g] + bv;
          if (Cadd) v += Cadd[(size_t)gm * N + gn];
          if (doRelu) v = fmaxf(v, 0.f);
          D[(size_t)gm * N + gn] = v;
        }
      }
    }
  }
}

// ---------------------------------------------------------------------------
// Utility / graph kernels
// ---------------------------------------------------------------------------
__global__ void fill_f32(float* p, float v, long long n) {
  long long i = (long long)blockIdx.x * 256 + threadIdx.x;
  long long stride = (long long)gridDim.x * 256;
  for (; i < n; i += stride) p[i] = v;
}

__device__ __forceinline__ unsigned fmono(float f) {
  unsigned u = __float_as_uint(f);
  return (u & 0x80000000u) ? ~u : (u | 0x80000000u);
}

// GATv2 pass A: logit[e,h] = sum_c lrelu(xl[s]+xr[d]) * att[h,c]; segment-max into mmono
__global__ __launch_bounds__(256) void gat_logit_kernel(
    const float* __restrict__ xl, const float* __restrict__ xr,
    const int* __restrict__ src, const int* __restrict__ dst,
    const float* __restrict__ att, float* __restrict__ logit,
    unsigned* __restrict__ mmono, int E, int E2)
{
  int wave = threadIdx.x >> 5, lane = threadIdx.x & 31;
  int e = blockIdx.x * 8 + wave;
  if (e >= E2) return;
  int s = (e < E) ? src[e] : (e - E);
  int d = (e < E) ? dst[e] : (e - E);
  float a0 = xl[s * 64 + lane]      + xr[d * 64 + lane];
  float a1 = xl[s * 64 + 32 + lane] + xr[d * 64 + 32 + lane];
  a0 = (a0 > 0.f) ? a0 : 0.2f * a0;        // leaky_relu(0.2)
  a1 = (a1 > 0.f) ? a1 : 0.2f * a1;
  float p0 = a0 * att[lane];               // head 0 (channels 0-31)
  float p1 = a1 * att[32 + lane];          // head 1 (channels 32-63)
  for (int off = 16; off > 0; off >>= 1) {
    p0 += __shfl_xor(p0, off, 32);
    p1 += __shfl_xor(p1, off, 32);
  }
  if (lane == 0) {
    logit[e * 2 + 0] = p0;
    logit[e * 2 + 1] = p1;
    atomicMax(&mmono[d * 2 + 0], fmono(p0));
    atomicMax(&mmono[d * 2 + 1], fmono(p1));
  }
}

__global__ void m_final_kernel(const unsigned* mmono, float* mf, int n) {
  int i = blockIdx.x * 256 + threadIdx.x;
  if (i >= n) return;
  unsigned u = mmono[i];
  unsigned bits = (u & 0x80000000u) ? (u ^ 0x80000000u) : ~u;
  float f = __uint_as_float(bits);
  if ((bits & 0x7F800000u) == 0x7F800000u) f = 0.f;  // non-finite -> 0
  mf[i] = f;
}

__global__ void gat_expsum_kernel(float* logit, const float* mf,
    const int* __restrict__ src, const int* __restrict__ dst,
    float* ssum, int E, int E2)
{
  int e = blockIdx.x * 256 + threadIdx.x;
  if (e >= E2) return;
  int d = (e < E) ? dst[e] : (e - E);
  float e0 = expf(logit[e * 2 + 0] - mf[d * 2 + 0]);
  float e1 = expf(logit[e * 2 + 1] - mf[d * 2 + 1]);
  logit[e * 2 + 0] = e0;
  logit[e * 2 + 1] = e1;
  atomicAdd(&ssum[d * 2 + 0], e0);
  atomicAdd(&ssum[d * 2 + 1], e1);
}

__global__ __launch_bounds__(256) void gat_scatter_kernel(
    const float* __restrict__ logit, const float* __restrict__ ssum,
    const float* __restrict__ xl,
    const int* __restrict__ src, const int* __restrict__ dst,
    float* __restrict__ out, int E, int E2)
{
  int wave = threadIdx.x >> 5, lane = threadIdx.x & 31;
  int e = blockIdx.x * 8 + wave;
  if (e >= E2) return;
  int s = (e < E) ? src[e] : (e - E);
  int d = (e < E) ? dst[e] : (e - E);
  float a0 = logit[e * 2 + 0] / (ssum[d * 2 + 0] + 1e-16f);
  float a1 = logit[e * 2 + 1] / (ssum[d * 2 + 1] + 1e-16f);
  atomicAdd(&out[d * 64 + lane],      a0 * xl[s * 64 + lane]);
  atomicAdd(&out[d * 64 + 32 + lane], a1 * xl[s * 64 + 32 + lane]);
}

// PairNorm pieces (conv bias cancels under column-centering, so it is skipped)
__global__ void colsum_kernel(const float* x, float* colsum, int nrows) {
  int j = blockIdx.x * 256 + threadIdx.x;
  int col = j & 63;
  int r0 = j >> 6;
  int rstride = (gridDim.x * 256) >> 6;
  float acc = 0.f;
  for (int r = r0; r < nrows; r += rstride) acc += x[r * 64 + col];
  atomicAdd(&colsum[col], acc);
}

__global__ void sumsq_kernel(const float* x, const float* colsum, float* sumsq, int nrows) {
  long long n = (long long)nrows * 64;
  float invN = 1.f / (float)nrows;
  float acc = 0.f;
  for (long long i = (long long)blockIdx.x * 256 + threadIdx.x; i < n;
       i += (long long)gridDim.x * 256) {
    float v = x[i] - colsum[(int)(i & 63)] * invN;
    acc += v * v;
  }
  for (int off = 16; off > 0; off >>= 1) acc += __shfl_xor(acc, off, 32);
  if ((threadIdx.x & 31) == 0) atomicAdd(sumsq, acc);
}

__global__ void pairnorm_kernel(const float* x, const float* colsum,
                                const float* sumsq, float* out, int nrows) {
  int i = blockIdx.x * 256 + threadIdx.x;
  if (i >= nrows * 64) return;
  float invN = 1.f / (float)nrows;
  float rs = rsqrtf(1e-5f + sumsq[0] * invN);
  float v = (x[i] - colsum[i & 63] * invN) * rs;
  out[i] = fmaxf(v, 0.f);   // relu(pairnorm(...))
}

// ef[e] = [h1[src] | h2[src] | h1[dst] | h2[dst]]  (jump-knowledge cat + src/dst cat)
__global__ __launch_bounds__(256) void edge_concat_kernel(
    const float* __restrict__ h1, const float* __restrict__ h2,
    const int* __restrict__ src, const int* __restrict__ dst,
    float* __restrict__ ef)
{
  int e = blockIdx.x, t = threadIdx.x;
  int s = src[e], d = dst[e];
  int c = t & 63;
  const float* base;
  int node;
  switch (t >> 6) {
    case 0:  base = h1; node = s; break;
    case 1:  base = h2; node = s; break;
    case 2:  base = h1; node = d; break;
    default: base = h2; node = d; break;
  }
  ef[(size_t)e * 256 + t] = base[node * 64 + c];
}

__global__ __launch_bounds__(256) void agg_scatter_kernel(
    const float* __restrict__ ef, const int* __restrict__ src,
    const int* __restrict__ dst, float* __restrict__ agg)
{
  int e = blockIdx.x, t = threadIdx.x;
  float v = ef[(size_t)e * 256 + t];
  atomicAdd(&agg[(size_t)src[e] * 256 + t], v);
  atomicAdd(&agg[(size_t)dst[e] * 256 + t], v);
}

// Wcomb = W0[0:256] - W0[256:512] - W0[512:768]   (rows of mp_W0)
__global__ void wcomb_kernel(const float* W0, float* Wc) {
  int i = blockIdx.x * 256 + threadIdx.x;
  if (i < 256 * 256) Wc[i] = W0[i] - W0[i + 65536] - W0[i + 131072];
}

// hid[e] = relu(hid[e] + tmpb[src[e]] + tmpc[dst[e]])
__global__ __launch_bounds__(256) void hid_epi_kernel(
    float* __restrict__ hid, const float* __restrict__ tb,
    const float* __restrict__ tc, const int* __restrict__ src,
    const int* __restrict__ dst)
{
  int e = blockIdx.x, t = threadIdx.x;
  size_t idx = (size_t)e * 256 + t;
  float v = hid[idx] + tb[(size_t)src[e] * 256 + t] + tc[(size_t)dst[e] * 256 + t];
  hid[idx] = fmaxf(v, 0.f);
}

// ---------------------------------------------------------------------------
extern "C" void kernel_launch(void* const* d_in, const int* in_sizes, int n_in,
                              void* d_out, int out_size, void* d_ws, size_t ws_size,
                              hipStream_t stream)
{
  const float* x      = (const float*)d_in[0];
  const int*   ei     = (const int*)  d_in[1];
  const float* c0_Wl  = (const float*)d_in[2];
  const float* c0_bl  = (const float*)d_in[3];
  const float* c0_Wr  = (const float*)d_in[4];
  const float* c0_br  = (const float*)d_in[5];
  const float* c0_att = (const float*)d_in[6];
  // d_in[7] conv0_b: cancels in PairNorm centering -> unused
  const float* c1_Wl  = (const float*)d_in[8];
  const float* c1_bl  = (const float*)d_in[9];
  const float* c1_Wr  = (const float*)d_in[10];
  const float* c1_br  = (const float*)d_in[11];
  const float* c1_att = (const float*)d_in[12];
  // d_in[13] conv1_b: unused (same reason)
  const float* mp_W0  = (const float*)d_in[14];
  const float* mp_b0  = (const float*)d_in[15];
  const float* mp_W1  = (const float*)d_in[16];
  const float* mp_b1  = (const float*)d_in[17];
  const float* fc0_W  = (const float*)d_in[18];
  const float* fc0_b  = (const float*)d_in[19];
  const float* fc1_W  = (const float*)d_in[20];
  const float* fc1_b  = (const float*)d_in[21];
  const float* out_W  = (const float*)d_in[22];
  const float* out_b  = (const float*)d_in[23];

  const int Nn = in_sizes[0] / 64;
  const int E  = in_sizes[1] / 2;
  const int E2 = E + Nn;                 // edges + self loops
  const int* src = ei;
  const int* dst = ei + E;

  float* base = (float*)d_ws;
  size_t off = 0;
  auto alloc = [&](size_t nf) {
    float* p = base + off;
    off += (nf + 63) & ~(size_t)63;      // 256B-aligned slabs
    return p;
  };
  float* ef     = alloc((size_t)E * 256);
  float* hid    = alloc((size_t)E * 256);
  float* logit  = alloc((size_t)E2 * 2);
  float* xl     = alloc((size_t)Nn * 64);
  float* xr     = alloc((size_t)Nn * 64);
  float* h1     = alloc((size_t)Nn * 64);
  float* h2     = alloc((size_t)Nn * 64);
  float* gat    = alloc((size_t)Nn * 64);
  float* mmono  = alloc((size_t)Nn * 2);
  float* mf     = alloc((size_t)Nn * 2);
  float* ssum   = alloc((size_t)Nn * 2);
  float* colsum = alloc(64);
  float* sumsq  = alloc(64);
  float* agg    = alloc((size_t)Nn * 256);
  float* tmpb   = alloc((size_t)Nn * 256);
  float* tmpc   = alloc((size_t)Nn * 256);
  float* wcomb  = alloc((size_t)256 * 256);
  (void)ws_size; (void)n_in; (void)out_size;

  auto gemm = [&](const float* A, const float* B, const float* bias,
                  const float* Cadd, float* D, int M, int N, int K, int relu) {
    dim3 g((unsigned)((N + 63) / 64), (unsigned)((M + 127) / 128));
    gemm_wmma_f32<<<g, 256, 0, stream>>>(A, B, bias, Cadd, D, M, N, K, relu);
  };
  auto fill = [&](float* p, float v, long long n) {
    long long b = (n + 255) / 256; if (b > 1024) b = 1024;
    fill_f32<<<(unsigned)b, 256, 0, stream>>>(p, v, n);
  };

  auto conv = [&](const float* in, const float* Wl, const float* bl,
                  const float* Wr, const float* br, const float* att, float* outH) {
    gemm(in, Wl, bl, nullptr, xl, Nn, 64, 64, 0);
    gemm(in, Wr, br, nullptr, xr, Nn, 64, 64, 0);
    fill(mmono, 0.f, (long long)Nn * 2);     // 0 == floor of monotone encoding
    fill(ssum,  0.f, (long long)Nn * 2);
    fill(gat,   0.f, (long long)Nn * 64);
    fill(colsum, 0.f, 64);
    fill(sumsq,  0.f, 1);
    gat_logit_kernel<<<(E2 + 7) / 8, 256, 0, stream>>>(
        xl, xr, src, dst, att, logit, (unsigned*)mmono, E, E2);
    m_final_kernel<<<(2 * Nn + 255) / 256, 256, 0, stream>>>(
        (const unsigned*)mmono, mf, 2 * Nn);
    gat_expsum_kernel<<<(E2 + 255) / 256, 256, 0, stream>>>(
        logit, mf, src, dst, ssum, E, E2);
    gat_scatter_kernel<<<(E2 + 7) / 8, 256, 0, stream>>>(
        logit, ssum, xl, src, dst, gat, E, E2);
    colsum_kernel<<<256, 256, 0, stream>>>(gat, colsum, Nn);
    sumsq_kernel<<<256, 256, 0, stream>>>(gat, colsum, sumsq, Nn);
    pairnorm_kernel<<<(Nn * 64 + 255) / 256, 256, 0, stream>>>(
        gat, colsum, sumsq, outH, Nn);
  };

  // Two GATv2 + PairNorm + ReLU layers
  conv(x,  c0_Wl, c0_bl, c0_Wr, c0_br, c0_att, h1);
  conv(h1, c1_Wl, c1_bl, c1_Wr, c1_br, c1_att, h2);

  // Edge features + neighborhood aggregation
  edge_concat_kernel<<<E, 256, 0, stream>>>(h1, h2, src, dst, ef);
  fill(agg, 0.f, (long long)Nn * 256);
  agg_scatter_kernel<<<E, 256, 0, stream>>>(ef, src, dst, agg);

  // msg @ W0 decomposition: ef@(W0a-W0b-W0c) + agg[src]@W0b + agg[dst]@W0c
  wcomb_kernel<<<256, 256, 0, stream>>>(mp_W0, wcomb);
  gemm(agg, mp_W0 + 65536,  nullptr, nullptr, tmpb, Nn, 256, 256, 0);
  gemm(agg, mp_W0 + 131072, nullptr, nullptr, tmpc, Nn, 256, 256, 0);
  gemm(ef, wcomb, mp_b0, nullptr, hid, E, 256, 256, 0);
  hid_epi_kernel<<<E, 256, 0, stream>>>(hid, tmpb, tmpc, src, dst);

  // Residual MP output, then FC head (in-place C-accumulate is tile-safe)
  gemm(hid, mp_W1, mp_b1, ef, ef, E, 256, 256, 0);          // ef = hid@W1 + b1 + ef
  gemm(ef,  fc0_W, fc0_b, nullptr, hid, E, 128, 256, 1);    // e0 = relu(...)  (reuse hid)
  gemm(hid, fc1_W, fc1_b, nullptr, ef,  E,  64, 128, 1);    // e1 = relu(...)  (reuse ef)
  gemm(ef,  out_W, out_b, nullptr, (float*)d_out, E, 8, 64, 0);
}